// Calculating_COG_66623532696119
// MI455X (gfx1250) — compile-verified
//
#include <hip/hip_runtime.h>
#include <math.h>

#define NSIDE 256
#define N_FACETS (12 * NSIDE * NSIDE)
#define TPB 256
#define TILES 4   // parents per block = TPB*TILES

// ---------------------------------------------------------------------------
// HEALPix nested pix -> unit vector, nside=256 fixed (matches reference f32 math)
// ---------------------------------------------------------------------------
__device__ __forceinline__ unsigned compress_bits(unsigned v) {
    v &= 0x55555555u;
    v = (v | (v >> 1)) & 0x33333333u;
    v = (v | (v >> 2)) & 0x0F0F0F0Fu;
    v = (v | (v >> 4)) & 0x00FF00FFu;
    v = (v | (v >> 8)) & 0x0000FFFFu;
    return v;
}

__device__ __forceinline__ void pix2vec_nest256(int pix, float& X, float& Y, float& Z) {
    const int nside = NSIDE;
    int face = pix >> 16;                        // npface = 65536
    unsigned ipf = (unsigned)pix & 0xFFFFu;
    int ix = (int)compress_bits(ipf);
    int iy = (int)compress_bits(ipf >> 1);
    int jrll = 2 + (face >> 2);                               // {2,2,2,2,3,3,3,3,4,4,4,4}
    int jpll = 2 * (face & 3) + (((face >> 2) == 1) ? 0 : 1); // {1,3,5,7,0,2,4,6,1,3,5,7}
    int jr = jrll * nside - ix - iy - 1;
    bool north = jr < nside;
    bool south = jr > 3 * nside;
    int nr = north ? jr : (south ? (4 * nside - jr) : nside);
    const float fact2 = 4.0f / (12.0f * (float)nside * (float)nside);
    const float fact1 = (2.0f * (float)nside) * fact2;
    float nrf = (float)nr;
    float z = north ? (1.0f - nrf * nrf * fact2)
                    : (south ? (nrf * nrf * fact2 - 1.0f)
                             : ((float)(2 * nside - jr) * fact1));
    int kshift = (north || south) ? 0 : ((jr - nside) & 1);
    int num = jpll * nr + ix - iy + 1 + kshift;
    int jp = num / 2;                            // C trunc division == reference
    if (jp > 4 * nr) jp -= 4 * nr;
    if (jp < 1)      jp += 4 * nr;
    float phi = ((float)jp - ((float)kshift + 1.0f) * 0.5f) * 1.57079632679489662f / nrf;
    float sth = sqrtf((1.0f - z) * (1.0f + z));
    float s, c;
    sincosf(phi, &s, &c);
    X = sth * c;
    Y = sth * s;
    Z = z;
}

// ---------------------------------------------------------------------------
// Kernel 1: zero the per-facet {x,y,z,score} table (score==0 encodes non-member)
// ---------------------------------------------------------------------------
__global__ void zero_table_kernel(float4* __restrict__ table, int n) {
    int i = blockIdx.x * blockDim.x + threadIdx.x;
    if (i < n) table[i] = make_float4(0.f, 0.f, 0.f, 0.f);
}

// ---------------------------------------------------------------------------
// Kernel 2: scatter-build table[f0] = {pix2vec(a), score}  (f0 unique -> no races)
// Replicates the reference's lut_corr[0]=0 and lut_kpt[0]=0 overrides.
// ---------------------------------------------------------------------------
__global__ void build_table_kernel(const int* __restrict__ corr,
                                   const int* __restrict__ f0s,
                                   const int* __restrict__ f1s,
                                   const float* __restrict__ scores1,
                                   float4* __restrict__ table, int n0) {
    int i = blockIdx.x * blockDim.x + threadIdx.x;
    if (i >= n0) return;
    int f0 = f0s[i];
    int k1 = corr[i];
    int a  = (f0 == 0) ? 0 : f1s[k1];   // lut_corr with [0]=0 override
    int b  = (a == 0) ? 0 : k1;         // lut_kpt with [0]=0 override
    float s = scores1[b];
    float x, y, z;
    pix2vec_nest256(a, x, y, z);
    table[f0] = make_float4(x, y, z, s);
}

// ---------------------------------------------------------------------------
// Kernel 3: one thread per parent, TILES tiles per block. The streaming filt
// read is software-pipelined through double-buffered LDS using the CDNA5
// async global->LDS DMA: ASYNCcnt is an in-order per-wave FIFO, so
// s_wait_asynccnt 0x1 retires exactly the oldest outstanding tile while the
// next one streams in behind the table gathers + CoG math.
// ---------------------------------------------------------------------------
__device__ __forceinline__ void async_load16(unsigned lds_addr, const int* gptr) {
    unsigned long long ga = (unsigned long long)(size_t)gptr;
    asm volatile("global_load_async_to_lds_b128 %0, %1, off"
                 :: "v"(lds_addr), "v"(ga) : "memory");
}

__global__ void __launch_bounds__(TPB)
cog_kernel(const int* __restrict__ filt,
           const float4* __restrict__ table,
           float* __restrict__ pos_out,     // [P,3]
           float* __restrict__ score_out,   // [P]
           int P) {
    __shared__ int4 sh[2][TPB];
    const int tid  = threadIdx.x;
    const int base = blockIdx.x * (TPB * TILES) + tid;   // parent index, tile 0

    // LDS byte addresses: ISA aperture rule -> low 32 bits of flat LDS pointer.
    unsigned lds_buf[2];
    lds_buf[0] = (unsigned)(size_t)(&sh[0][tid]);
    lds_buf[1] = (unsigned)(size_t)(&sh[1][tid]);

    // Prime the pipeline: DMA tile 0.
    if (base < P) async_load16(lds_buf[0], filt + 4 * (size_t)base);

    for (int t = 0; t < TILES; ++t) {
        int p = base + t * TPB;

        // Issue DMA for tile t+1 into the other buffer, then retire tile t.
        int pn = p + TPB;
        if (t + 1 < TILES && pn < P)
            async_load16(lds_buf[(t + 1) & 1], filt + 4 * (size_t)pn);
        if (t + 1 < TILES) {
            asm volatile("s_wait_asynccnt 0x1" ::: "memory");
        } else {
            asm volatile("s_wait_asynccnt 0x0" ::: "memory");
        }

        if (p >= P) break;
        int4 f = sh[t & 1][tid];

        float4 t0 = table[f.x];
        float4 t1 = table[f.y];
        float4 t2 = table[f.z];
        float4 t3 = table[f.w];

        float den = t0.w + t1.w + t2.w + t3.w;
        float nx = t0.x * t0.w + t1.x * t1.w + t2.x * t2.w + t3.x * t3.w;
        float ny = t0.y * t0.w + t1.y * t1.w + t2.y * t2.w + t3.y * t3.w;
        float nz = t0.z * t0.w + t1.z * t1.w + t2.z * t2.w + t3.z * t3.w;
        int cnt = (t0.w > 0.f) + (t1.w > 0.f) + (t2.w > 0.f) + (t3.w > 0.f);

        float px = (den == 0.f) ? 0.f : nx / den;
        float py = (den == 0.f) ? 0.f : ny / den;
        float pz = (den == 0.f) ? 0.f : nz / den;
        float scv = (cnt == 0) ? 0.f : den / (float)cnt;

        size_t o = 3 * (size_t)p;
        pos_out[o + 0] = px;
        pos_out[o + 1] = py;
        pos_out[o + 2] = pz;
        score_out[p] = scv;
    }
}

// ---------------------------------------------------------------------------
// Kernel 4: trivial tail outputs (arange, correspondences, score copy)
// ---------------------------------------------------------------------------
__global__ void tail_kernel(const int* __restrict__ corr,
                            const float* __restrict__ scores1,
                            int* __restrict__ out_k0,
                            int* __restrict__ out_k1,
                            float* __restrict__ out_s,
                            int n0, int n1) {
    int i = blockIdx.x * blockDim.x + threadIdx.x;
    if (i < n0) {
        out_k0[i] = i;
        out_k1[i] = corr[i];
    }
    if (i < n1) {
        out_s[i] = scores1[i];
    }
}

extern "C" void kernel_launch(void* const* d_in, const int* in_sizes, int n_in,
                              void* d_out, int out_size, void* d_ws, size_t ws_size,
                              hipStream_t stream) {
    (void)n_in; (void)out_size; (void)ws_size;
    // setup_inputs order:
    // 0: nside(1)  1: correspondences[N0]  2: img0_child_facets[N0]
    // 3: img1_child_facets[N1]  4: filt[M]  5: keypointScores1[N1]
    const int*   corr = (const int*)d_in[1];
    const int*   f0s  = (const int*)d_in[2];
    const int*   f1s  = (const int*)d_in[3];
    const int*   filt = (const int*)d_in[4];
    const float* sc   = (const float*)d_in[5];
    int n0 = in_sizes[1];
    int n1 = in_sizes[5];
    int M  = in_sizes[4];
    int P  = M / 4;

    float4* table = (float4*)d_ws;   // N_FACETS * 16B = 12.58 MB scratch

    // Output layout (flat, return order): pos_cog[3P] | score_cog[P] | k0[N0] | k1[N0] | scores[N1]
    float* out       = (float*)d_out;
    float* pos_out   = out;
    float* score_out = out + 3 * (size_t)P;
    int*   k0_out    = (int*)(out + 4 * (size_t)P);
    int*   k1_out    = k0_out + n0;
    float* s_out     = (float*)(k1_out + n0);

    zero_table_kernel<<<(N_FACETS + 255) / 256, 256, 0, stream>>>(table, N_FACETS);
    build_table_kernel<<<(n0 + 255) / 256, 256, 0, stream>>>(corr, f0s, f1s, sc, table, n0);
    int cog_blocks = (P + TPB * TILES - 1) / (TPB * TILES);
    cog_kernel<<<cog_blocks, TPB, 0, stream>>>(filt, table, pos_out, score_out, P);
    int nt = (n0 > n1) ? n0 : n1;
    tail_kernel<<<(nt + 255) / 256, 256, 0, stream>>>(corr, sc, k0_out, k1_out, s_out, n0, n1);
}